// S4DKernel_73538430042245
// MI455X (gfx1250) — compile-verified
//
#include <hip/hip_runtime.h>
#include <hip/hip_bf16.h>

// S4D ZOH kernel generation, recast as per-head 64x64x128 fp32 GEMM so the
// mode reduction runs on V_WMMA_F32_16X16X4_F32. Memory-bound (16MB out,
// ~0.7us HBM floor @ 23.3TB/s): fp32 WMMA costs nothing vs f16 and keeps
// full precision vs the complex64 reference. Epilogue stages the 64x128
// tile in LDS and drains it with CDNA5 async LDS->global B128 stores
// (512B contiguous per wave-instruction, no VGPR data movement).

typedef __attribute__((ext_vector_type(2))) float v2f;
typedef __attribute__((ext_vector_type(8))) float v8f;

#define S4D_H      512
#define S4D_N      32      // NHALF complex modes
#define S4D_L      8192
#define S4D_B      128     // l1 chunk width
#define S4D_L0     64      // S4D_L / S4D_B
#define S4D_K      64      // 2*N (re/im split inner dim)
#define SA_PAD     66      // 64 + 2 pad: row stride 264B breaks 256B bank period
#define SO_PAD     132     // 128 + 4 pad: rows r and r+8 hit disjoint bank sets

union SmemB {
  float B[S4D_K][S4D_B];       // GEMM B operand (64 x 128)
  float Out[S4D_L0][SO_PAD];   // staged output tile (64 x 128, padded)
};

__global__ __launch_bounds__(256) void s4d_wmma_kernel(
    const float* __restrict__ Cin,        // (H, N, 2)
    const float* __restrict__ log_dt,     // (H,)
    const float* __restrict__ log_A_real, // (H, N)
    const float* __restrict__ A_imag,     // (H, N)
    float* __restrict__ out)              // (H, L)
{
  __shared__ float s_dre[S4D_N], s_dim[S4D_N];   // dtA per mode
  __shared__ float s_cr[S4D_N],  s_ci[S4D_N];    // Cs = C*(e^dtA - 1)/A
  __shared__ float sA[S4D_L0][SA_PAD];           // 64 x 64 (padded)
  __shared__ SmemB sm;

  const int h   = blockIdx.x;
  const int tid = threadIdx.x;

  // ---- Phase 1: per-mode coefficients (32 threads) -----------------------
  if (tid < S4D_N) {
    const int n = tid;
    const float dt = expf(log_dt[h]);
    const float ar = -expf(log_A_real[h * S4D_N + n]);  // Re(A)
    const float ai = A_imag[h * S4D_N + n];             // Im(A)
    const float dre = ar * dt;                          // Re(dtA)
    const float dim = ai * dt;                          // Im(dtA)
    // e = exp(dtA)
    float es, ec;
    sincosf(dim, &es, &ec);
    const float em = expf(dre);
    const float nre = em * ec - 1.0f;                   // e - 1 (real)
    const float nim = em * es;                          // e - 1 (imag)
    // q = (e-1)/A = (e-1)*conj(A)/|A|^2
    const float inv = 1.0f / (ar * ar + ai * ai);
    const float qre = (nre * ar + nim * ai) * inv;
    const float qim = (nim * ar - nre * ai) * inv;
    // Cs = Cc * q
    const float c0 = Cin[(h * S4D_N + n) * 2 + 0];
    const float c1 = Cin[(h * S4D_N + n) * 2 + 1];
    s_dre[n] = dre;
    s_dim[n] = dim;
    s_cr[n]  = c0 * qre - c1 * qim;
    s_ci[n]  = c0 * qim + c1 * qre;
  }
  __syncthreads();

  // ---- Phase 2a: A matrix: rows l0, cols k = [2*Re(Cs*zB^l0) | -2*Im(..)] -
  for (int i = tid; i < S4D_L0 * S4D_K; i += 256) {
    const int l0 = i >> 6;
    const int k  = i & 63;
    const int n  = k & (S4D_N - 1);
    const float t0 = (float)(l0 * S4D_B);
    float s, c;
    sincosf(s_dim[n] * t0, &s, &c);
    const float mag = expf(s_dre[n] * t0);
    const float ure = mag * c, uim = mag * s;           // zB^l0
    const float re = s_cr[n] * ure - s_ci[n] * uim;     // Re(Cs*zB^l0)
    const float im = s_cr[n] * uim + s_ci[n] * ure;     // Im(Cs*zB^l0)
    sA[l0][k] = (k < S4D_N) ? (2.0f * re) : (-2.0f * im);
  }

  // ---- Phase 2b: B matrix: rows k = [Re(z^l1) | Im(z^l1)], cols l1 -------
  for (int i = tid; i < S4D_K * S4D_B; i += 256) {
    const int k  = i >> 7;
    const int l1 = i & (S4D_B - 1);
    const int n  = k & (S4D_N - 1);
    const float t = (float)l1;
    float s, c;
    sincosf(s_dim[n] * t, &s, &c);
    const float mag = expf(s_dre[n] * t);
    sm.B[k][l1] = (k < S4D_N) ? (mag * c) : (mag * s);
  }
  __syncthreads();

  // ---- Phase 3: 64x128 = 4x8 tiles of 16x16; 8 waves, 4 tiles each -------
  // Scalarize the wave id so the tile loop is uniform (no EXEC masking
  // around WMMA; ISA requires EXEC all-1s).
  const int wave = __builtin_amdgcn_readfirstlane(tid >> 5);
  const int lane = tid & 31;
  const int mn   = lane & 15;            // row-of-A / col-of-B within tile
  const int koff = (lane >> 4) << 1;     // K pair select: 0 or 2
  const int radd = (lane >> 4) << 3;     // C/D row add for upper lanes

  v8f acc[4];
#pragma unroll
  for (int i = 0; i < 4; ++i) {
    const int t  = wave + (i << 3);      // uniform tile id
    const int tm = (t >> 3) << 4;        // 0,16,32,48
    const int tn = (t & 7) << 4;         // 0..112
    v8f c = {};
#pragma unroll
    for (int kk = 0; kk < S4D_K; kk += 4) {
      // A 16x4: lane holds row (tm+mn), K = {kk+koff, kk+koff+1}
      v2f a = *(const v2f*)&sA[tm + mn][kk + koff];
      // B 4x16: lane holds col (tn+mn), rows K = {kk+koff, kk+koff+1}
      v2f b;
      b.x = sm.B[kk + koff + 0][tn + mn];
      b.y = sm.B[kk + koff + 1][tn + mn];
      c = __builtin_amdgcn_wmma_f32_16x16x4_f32(
          /*neg_a=*/false, a, /*neg_b=*/false, b,
          /*c_mod=*/(short)0, c, /*reuse_a=*/false, /*reuse_b=*/false);
    }
    acc[i] = c;
  }

  // All waves done reading sm.B before we overwrite it as sm.Out.
  __syncthreads();

  // ---- Phase 4: stage D tiles into LDS (padded) --------------------------
#pragma unroll
  for (int i = 0; i < 4; ++i) {
    const int t  = wave + (i << 3);
    const int tm = (t >> 3) << 4;
    const int tn = (t & 7) << 4;
#pragma unroll
    for (int r = 0; r < 8; ++r) {
      sm.Out[tm + r + radd][tn + mn] = acc[i][r];
    }
  }
  __syncthreads();

  // ---- Phase 5: async LDS -> global drain, 512B contiguous per wave-op ---
  // 64*128 floats = 2048 b128 chunks; 8 per thread. The s_wait_asynccnt 0
  // at the end drains ASYNCcnt; LDS stays live until the wave ends.
  float* const gbase = out + (size_t)h * S4D_L;
#pragma unroll
  for (int j = 0; j < 8; ++j) {
    const int lin = (j * 256 + tid) * 4;          // float index into K[h,:]
    const int row = lin >> 7;
    const int col = lin & (S4D_B - 1);
    const unsigned lds_off = (unsigned)(uintptr_t)(&sm.Out[row][col]);
    float* gdst = gbase + lin;
    asm volatile("global_store_async_from_lds_b128 %0, %1, off"
                 :: "v"(gdst), "v"(lds_off)
                 : "memory");
  }
  asm volatile("s_wait_asynccnt 0x0" ::: "memory");
}

extern "C" void kernel_launch(void* const* d_in, const int* in_sizes, int n_in,
                              void* d_out, int out_size, void* d_ws, size_t ws_size,
                              hipStream_t stream) {
  const float* C          = (const float*)d_in[0];
  const float* log_dt     = (const float*)d_in[1];
  const float* log_A_real = (const float*)d_in[2];
  const float* A_imag     = (const float*)d_in[3];
  float* out = (float*)d_out;
  const int H = in_sizes[1];  // log_dt has H elements
  s4d_wmma_kernel<<<dim3(H), dim3(256), 0, stream>>>(
      C, log_dt, log_A_real, A_imag, out);
}